// NonLinearSAGE_54400055771179
// MI455X (gfx1250) — compile-verified
//
#include <hip/hip_runtime.h>

#define N_NODES  1500000
#define N_EDGES  48000000
#define N_OUT    500000            // N_NODES / 3
#define TILE_EDGES 1024            // edges per tile per workgroup (256 thr * 4)
#define TILES_PER_WG 5             // 9375 blocks * 5 * 1024 = 48e6 exactly
#define CPOL_NT 1                  // temporal hint: non-temporal load

typedef int   v4i __attribute__((ext_vector_type(4)));
typedef float v4f __attribute__((ext_vector_type(4)));

typedef __attribute__((address_space(1))) v4i* gptr_v4i;   // global, typed
typedef __attribute__((address_space(3))) v4i* lptr_v4i;   // LDS, typed

// ---------------------------------------------------------------------------
// Kernel 1: zero the compacted aggregate buffer (d_ws, 2 MB, L2-resident).
// ---------------------------------------------------------------------------
__global__ __launch_bounds__(256) void zero_agg_kernel(float* __restrict__ agg) {
    int i = blockIdx.x * blockDim.x + threadIdx.x;   // [0, N_OUT/4)
    if (i < N_OUT / 4) {
        v4f z = {0.f, 0.f, 0.f, 0.f};
        *((v4f*)agg + i) = z;
    }
}

// ---------------------------------------------------------------------------
// Kernel 2: edge scatter with async global->LDS double buffering.
// Only edges with dst % 3 == 0 survive to the output, so aggregate into
// compacted agg[dst/3] (2 MB -> L2-resident). The 384 MB edge stream is
// pulled by the async engine (ASYNCcnt) with NT hints while the wave is
// busy/stalled on the random x-gathers and f32 atomics of the previous
// tile -- per-wave pipelining at zero VGPR cost.
// Each thread stages its own 16B dst + 16B src chunk; no barriers needed.
// ---------------------------------------------------------------------------
__global__ __launch_bounds__(256) void scatter_kernel(
        const int* __restrict__ src_,
        const int* __restrict__ dst_,
        const float* __restrict__ x,
        float* __restrict__ agg) {
    __shared__ int dbuf[2][TILE_EDGES];
    __shared__ int sbuf[2][TILE_EDGES];

    int* src = const_cast<int*>(src_);
    int* dst = const_cast<int*>(dst_);

    const int tid = threadIdx.x;
    const long long wgbase = (long long)blockIdx.x * (TILES_PER_WG * TILE_EDGES);
    const int lslot = tid * 4;

    // Prime the pipeline: issue tile 0 into buffer 0.
    {
        long long gb = wgbase + lslot;
        __builtin_amdgcn_global_load_async_to_lds_b128(
            (gptr_v4i)(dst + gb), (lptr_v4i)&dbuf[0][lslot], 0, CPOL_NT);
        __builtin_amdgcn_global_load_async_to_lds_b128(
            (gptr_v4i)(src + gb), (lptr_v4i)&sbuf[0][lslot], 0, CPOL_NT);
    }

#pragma unroll
    for (int i = 0; i < TILES_PER_WG; ++i) {
        if (i + 1 < TILES_PER_WG) {
            // Keep the HBM stream moving: prefetch tile i+1 into the other buffer.
            long long gb = wgbase + (long long)(i + 1) * TILE_EDGES + lslot;
            int nb = (i + 1) & 1;
            __builtin_amdgcn_global_load_async_to_lds_b128(
                (gptr_v4i)(dst + gb), (lptr_v4i)&dbuf[nb][lslot], 0, CPOL_NT);
            __builtin_amdgcn_global_load_async_to_lds_b128(
                (gptr_v4i)(src + gb), (lptr_v4i)&sbuf[nb][lslot], 0, CPOL_NT);
            __builtin_amdgcn_s_wait_asynccnt(2);   // tile i landed (in-order), i+1 in flight
        } else {
            __builtin_amdgcn_s_wait_asynccnt(0);   // drain for the last tile
        }

        const int b = i & 1;
        v4i d = *(const v4i*)&dbuf[b][lslot];       // ds_load_b128
        v4i s = *(const v4i*)&sbuf[b][lslot];

        unsigned int du[4] = {(unsigned)d.x, (unsigned)d.y, (unsigned)d.z, (unsigned)d.w};
        unsigned int su[4] = {(unsigned)s.x, (unsigned)s.y, (unsigned)s.z, (unsigned)s.w};

#pragma unroll
        for (int k = 0; k < 4; ++k) {
            unsigned int q = du[k] / 3u;           // mul-hi sequence, no hw divide
            if (q * 3u == du[k]) {                 // dst % 3 == 0 -> contributes
                float v = x[su[k]];                // random gather, L2-resident (6 MB)
                unsafeAtomicAdd(&agg[q], v);       // hw global_atomic_add_f32, no return
            }
        }
    }
}

// ---------------------------------------------------------------------------
// Kernel 3: fused SAGE combine + 1->2->1 MLP on the 500K kept nodes.
//   h   = agg[i]*w_l + x[3i]*w_r
//   out = W2[0]*relu(W1[0]*h+b1[0]) + W2[1]*relu(W1[1]*h+b1[1]) + b2
// ---------------------------------------------------------------------------
__global__ __launch_bounds__(256) void epilogue_kernel(
        const float* __restrict__ agg,
        const float* __restrict__ x,
        const float* __restrict__ w_l,
        const float* __restrict__ w_r,
        const float* __restrict__ W1,
        const float* __restrict__ b1,
        const float* __restrict__ W2,
        const float* __restrict__ b2,
        float* __restrict__ out) {
    int t = blockIdx.x * blockDim.x + threadIdx.x;   // [0, N_OUT/4)
    if (t >= N_OUT / 4) return;

    const float wl = w_l[0], wr = w_r[0];
    const float w10 = W1[0], w11 = W1[1];
    const float c10 = b1[0], c11 = b1[1];
    const float w20 = W2[0], w21 = W2[1];
    const float c2  = b2[0];

    v4f a  = *((const v4f*)agg + t);                 // agg[4t .. 4t+3]
    const v4f* xb = (const v4f*)(x + 12 * t);        // x[12t .. 12t+11], 48B-aligned
    v4f x0 = xb[0], x1 = xb[1], x2 = xb[2];
    float xr[4] = {x0.x, x0.w, x1.z, x2.y};          // x[3*(4t+j)]
    float av[4] = {a.x, a.y, a.z, a.w};

    v4f o;
    float ov[4];
#pragma unroll
    for (int j = 0; j < 4; ++j) {
        float h  = fmaf(av[j], wl, xr[j] * wr);
        float a0 = fmaxf(fmaf(h, w10, c10), 0.0f);
        float a1 = fmaxf(fmaf(h, w11, c11), 0.0f);
        ov[j] = fmaf(w20, a0, fmaf(w21, a1, c2));
    }
    o.x = ov[0]; o.y = ov[1]; o.z = ov[2]; o.w = ov[3];
    __builtin_nontemporal_store(o, (v4f*)out + t);   // write-once, keep out of cache
}

// ---------------------------------------------------------------------------
// Launch
// ---------------------------------------------------------------------------
extern "C" void kernel_launch(void* const* d_in, const int* in_sizes, int n_in,
                              void* d_out, int out_size, void* d_ws, size_t ws_size,
                              hipStream_t stream) {
    const float* x      = (const float*)d_in[0];
    const int*   edges  = (const int*)d_in[1];   // [2, N_EDGES] flat: src then dst
    const float* w_l    = (const float*)d_in[2];
    const float* w_r    = (const float*)d_in[3];
    const float* W1     = (const float*)d_in[4];
    const float* b1     = (const float*)d_in[5];
    const float* W2     = (const float*)d_in[6];
    const float* b2     = (const float*)d_in[7];
    float*       out    = (float*)d_out;

    const int* src = edges;
    const int* dst = edges + N_EDGES;

    float* agg = (float*)d_ws;   // N_OUT floats of scratch (2 MB)

    // 1) zero compacted accumulator (harness does not re-zero d_ws)
    {
        int threads = 256;
        int blocks  = (N_OUT / 4 + threads - 1) / threads;
        zero_agg_kernel<<<blocks, threads, 0, stream>>>(agg);
    }

    // 2) async double-buffered scatter: 9375 WGs x 5 tiles x 1024 edges
    {
        int threads = 256;
        int blocks  = N_EDGES / (TILES_PER_WG * TILE_EDGES);   // 9375
        scatter_kernel<<<blocks, threads, 0, stream>>>(src, dst, x, agg);
    }

    // 3) fused combine + MLP, 4 outputs/thread
    {
        int threads = 256;
        int blocks  = (N_OUT / 4 + threads - 1) / threads;
        epilogue_kernel<<<blocks, threads, 0, stream>>>(agg, x, w_l, w_r,
                                                        W1, b1, W2, b2, out);
    }
}